// AVIDSimilarityMemoryBank_59072980189428
// MI455X (gfx1250) — compile-verified
//
#include <hip/hip_runtime.h>

typedef __attribute__((ext_vector_type(2))) float v2f;
typedef __attribute__((ext_vector_type(8))) float v8f;

#define DIM      128
#define BSZ      256
#define KNEG     1024
#define MEMN     1000000
#define INV_TEMP (1.0f / 0.07f)

// ---------------------------------------------------------------------------
// Kernel A: row-wise L2 normalize (256 x 128) -> workspace
// ---------------------------------------------------------------------------
__global__ void avid_normalize_kernel(const float* __restrict__ emb,
                                      float* __restrict__ out) {
    __shared__ float red[DIM];
    const int b = blockIdx.x;
    const int t = threadIdx.x;           // 0..127
    const float x = emb[b * DIM + t];
    red[t] = x * x;
    __syncthreads();
    for (int s = 64; s > 0; s >>= 1) {
        if (t < s) red[t] += red[t + s];
        __syncthreads();
    }
    const float inv = 1.0f / fmaxf(sqrtf(red[0]), 1e-12f);
    out[b * DIM + t] = x * inv;
}

// ---------------------------------------------------------------------------
// Kernel B: negative scores via V_WMMA_F32_16X16X4_F32.
// Block = 256 threads = 8 waves, one block per batch row b.
// Each wave processes 16-k tiles: A = 16 gathered bank rows (16x4 K-slices),
// B = query vector broadcast to all 16 columns -> D columns all equal the
// 16 desired dot products (accumulated over d = 0..127 in steps of 4).
// ---------------------------------------------------------------------------
__global__ void avid_neg_scores_kernel(const int* __restrict__ y,
                                       const int* __restrict__ neg_idx,
                                       const float* __restrict__ view1,
                                       const float* __restrict__ view2,
                                       const float* __restrict__ vn,
                                       const float* __restrict__ an,
                                       float* __restrict__ out_scores) {
    __shared__ __align__(16) float sv[DIM];
    __shared__ __align__(16) float sa[DIM];

    const int b    = blockIdx.x;
    const int t    = threadIdx.x;        // 0..255
    const int lane = t & 31;
    const int wave = t >> 5;             // 0..7
    const int half = lane >> 4;          // 0 or 1 (K-pair selector per ISA layout)
    const int mrow = lane & 15;          // matrix row M this lane owns

    if (t < DIM) {
        sv[t] = vn[b * DIM + t];
        sa[t] = an[b * DIM + t];
    }
    __syncthreads();

    const int ypos = y[b];

    #pragma unroll
    for (int m = 0; m < 2; ++m) {
        // m==0: v2a -> a_neg rows (view2) dotted with v
        // m==1: a2v -> v_neg rows (view1) dotted with a
        const float* bank = (m == 0) ? view2 : view1;
        const float* svec = (m == 0) ? sv : sa;

        for (int tile = wave; tile < KNEG / 16; tile += 8) {
            const int k0  = tile * 16;
            int idx = neg_idx[b * KNEG + k0 + mrow];
            idx += (idx >= ypos) ? 1 : 0;    // shift past positive index

            // lane<16 reads K-elements {d,d+1} of its row; lane>=16 reads {d+2,d+3}
            const float* rowp = bank + (size_t)idx * DIM + 2 * half;
            const float* bvec = svec + 2 * half;

            v8f c = {0.f, 0.f, 0.f, 0.f, 0.f, 0.f, 0.f, 0.f};
            #pragma unroll
            for (int d = 0; d < DIM; d += 4) {
                const float2 af = *reinterpret_cast<const float2*>(rowp + d);
                const float2 bf = *reinterpret_cast<const float2*>(bvec + d);
                v2f A; A.x = af.x; A.y = af.y;
                v2f B; B.x = bf.x; B.y = bf.y;
                // D = A(16x4) * B(4x16) + C ; all 16 columns identical
                c = __builtin_amdgcn_wmma_f32_16x16x4_f32(
                        /*neg_a=*/false, A, /*neg_b=*/false, B,
                        /*c_mod=*/(short)0, c, /*reuse_a=*/false, /*reuse_b=*/false);
            }

            // Column N=0 of D: lane 0 holds M=0..7 in c[0..7], lane 16 holds M=8..15
            if (mrow == 0) {
                float* outp = out_scores + (size_t)m * BSZ * (KNEG + 1)
                                        + (size_t)b * (KNEG + 1)
                                        + 1 + k0 + half * 8;
                #pragma unroll
                for (int r = 0; r < 8; ++r) outp[r] = c[r] * INV_TEMP;
            }
        }
    }
}

// ---------------------------------------------------------------------------
// Kernel C: positive scores + momentum-normalized scatter into the output
// banks (which were pre-filled by a D2D copy earlier on the same stream).
// One block (128 threads) per batch row.
// ---------------------------------------------------------------------------
__global__ void avid_pos_update_kernel(const int* __restrict__ y,
                                       const float* __restrict__ view1,
                                       const float* __restrict__ view2,
                                       const float* __restrict__ vn,
                                       const float* __restrict__ an,
                                       float* __restrict__ out_scores,
                                       float* __restrict__ bank1,
                                       float* __restrict__ bank2) {
    __shared__ float red[DIM];
    const int b = blockIdx.x;
    const int t = threadIdx.x;           // 0..127
    const int row = y[b];

    const float vp = view1[(size_t)row * DIM + t];
    const float ap = view2[(size_t)row * DIM + t];
    const float vv = vn[b * DIM + t];
    const float aa = an[b * DIM + t];

    // dot(a_pos, v)
    red[t] = ap * vv; __syncthreads();
    for (int s = 64; s > 0; s >>= 1) { if (t < s) red[t] += red[t + s]; __syncthreads(); }
    const float d_av = red[0]; __syncthreads();

    // dot(v_pos, a)
    red[t] = vp * aa; __syncthreads();
    for (int s = 64; s > 0; s >>= 1) { if (t < s) red[t] += red[t + s]; __syncthreads(); }
    const float d_va = red[0]; __syncthreads();

    const float t1 = 0.5f * (vp + vv);
    const float t2 = 0.5f * (ap + aa);

    red[t] = t1 * t1; __syncthreads();
    for (int s = 64; s > 0; s >>= 1) { if (t < s) red[t] += red[t + s]; __syncthreads(); }
    const float n1 = red[0]; __syncthreads();

    red[t] = t2 * t2; __syncthreads();
    for (int s = 64; s > 0; s >>= 1) { if (t < s) red[t] += red[t + s]; __syncthreads(); }
    const float n2 = red[0];

    if (t == 0) {
        out_scores[(size_t)0 * BSZ * (KNEG + 1) + (size_t)b * (KNEG + 1)] = d_av * INV_TEMP;
        out_scores[(size_t)1 * BSZ * (KNEG + 1) + (size_t)b * (KNEG + 1)] = d_va * INV_TEMP;
    }
    bank1[(size_t)row * DIM + t] = t1 / fmaxf(sqrtf(n1), 1e-12f);
    bank2[(size_t)row * DIM + t] = t2 / fmaxf(sqrtf(n2), 1e-12f);
}

// ---------------------------------------------------------------------------
extern "C" void kernel_launch(void* const* d_in, const int* in_sizes, int n_in,
                              void* d_out, int out_size, void* d_ws, size_t ws_size,
                              hipStream_t stream) {
    const float* video = (const float*)d_in[0];
    const float* audio = (const float*)d_in[1];
    const int*   y     = (const int*)d_in[2];
    const int*   neg   = (const int*)d_in[3];
    const float* view1 = (const float*)d_in[4];
    const float* view2 = (const float*)d_in[5];

    float* out    = (float*)d_out;
    float* scores = out;                                        // [2, 256, 1025]
    float* bank1  = out + (size_t)2 * BSZ * (KNEG + 1);         // [1M, 128]
    float* bank2  = bank1 + (size_t)MEMN * DIM;                 // [1M, 128]

    float* vn = (float*)d_ws;                                   // normalized video
    float* an = vn + (size_t)BSZ * DIM;                         // normalized audio

    // 1) normalize queries into workspace
    avid_normalize_kernel<<<BSZ, DIM, 0, stream>>>(video, vn);
    avid_normalize_kernel<<<BSZ, DIM, 0, stream>>>(audio, an);

    // 2) gathered-dot scores via WMMA (reads original banks only)
    avid_neg_scores_kernel<<<BSZ, 256, 0, stream>>>(y, neg, view1, view2, vn, an, scores);

    // 3) bulk bank copy into d_out (dominant cost: ~2 GB of HBM traffic)
    hipMemcpyAsync(bank1, view1, (size_t)MEMN * DIM * sizeof(float),
                   hipMemcpyDeviceToDevice, stream);
    hipMemcpyAsync(bank2, view2, (size_t)MEMN * DIM * sizeof(float),
                   hipMemcpyDeviceToDevice, stream);

    // 4) positive scores + momentum scatter (after copies, same stream)
    avid_pos_update_kernel<<<BSZ, DIM, 0, stream>>>(y, view1, view2, vn, an,
                                                    scores, bank1, bank2);
}